// BWSGRicker_70188355551733
// MI455X (gfx1250) — compile-verified
//
#include <hip/hip_runtime.h>

// BWSG Ricker scan: strictly serial 10,000-step recurrence.
// One wave, lane 0 carries the state in VGPRs; all step-invariant
// exp(logpsi[*]) terms hoisted; output streamed with non-temporal stores.

#define T_STEPS    10000
#define BEAR_INTRO (10.0f / 400.0f)

__global__ __launch_bounds__(32)
void ricker_scan_kernel(const float* __restrict__ y0,
                        const float* __restrict__ params,
                        const float* __restrict__ logpsi,
                        float* __restrict__ out) {
    // gfx1250 global_prefetch_b8 on the (tiny) parameter blocks
    __builtin_prefetch(y0,     0, 3);
    __builtin_prefetch(params, 0, 3);
    __builtin_prefetch(logpsi, 0, 3);

    if (threadIdx.x != 0) return;   // serial recurrence: one lane only

    float B = y0[0];
    float W = y0[1];
    float S = y0[2];
    float G = y0[3];
    const float i_flag         = y0[4];
    const float interventional = (i_flag != 0.0f) ? 1.0f : 0.0f;
    const float jstar          = 10.0f + i_flag - 1.0f;   // intro step (float-exact)

    const float p0 = params[0];
    const float p1 = params[1];
    const float p2 = params[2];

    // Hoist all step-invariant transcendentals out of the loop.
    const float a0 = expf(logpsi[0]);   // growth rate of G
    const float l1 = logpsi[1];         // used raw inside exp(l1*S)
    const float c2 = expf(logpsi[2]);
    const float c3 = expf(logpsi[3]);
    const float c4 = expf(logpsi[4]);
    const float c5 = expf(logpsi[5]);
    const float c6 = expf(logpsi[6]);

    // Row 0: raw y0[:4] (unclamped), plus the zero pad column.
    __builtin_nontemporal_store(B,    out + 0);
    __builtin_nontemporal_store(W,    out + 1);
    __builtin_nontemporal_store(S,    out + 2);
    __builtin_nontemporal_store(G,    out + 3);
    __builtin_nontemporal_store(0.0f, out + 4);

    float* __restrict__ row = out + 5;
    for (int j = 1; j < T_STEPS; ++j, row += 5) {
        const float mask = interventional * (((float)j == jstar) ? 1.0f : 0.0f);

        // All updates read the OLD state (matches lax.scan step).
        const float nG = fminf(G * expf(a0 * (1.0f - G) - expf(l1 * S)), 1.0f);
        const float nS = fminf(S * expf(p0 * G * (1.0f - S) - c2 * (W + B) - c3), 1.0f);
        const float nW = fminf(W * expf(p1 * S * (1.0f - W) - c4 * B - c5), 1.0f);
        const float nB = fminf(mask * BEAR_INTRO
                               + B * expf(p2 * (S + W) * (1.0f - B) - c6), 1.0f);

        // Streamed write-once output: non-temporal (TH=NT) stores.
        __builtin_nontemporal_store(nB,   row + 0);
        __builtin_nontemporal_store(nW,   row + 1);
        __builtin_nontemporal_store(nS,   row + 2);
        __builtin_nontemporal_store(nG,   row + 3);
        __builtin_nontemporal_store(0.0f, row + 4);

        B = nB; W = nW; S = nS; G = nG;
    }
}

extern "C" void kernel_launch(void* const* d_in, const int* in_sizes, int n_in,
                              void* d_out, int out_size, void* d_ws, size_t ws_size,
                              hipStream_t stream) {
    const float* y0     = (const float*)d_in[0];
    const float* params = (const float*)d_in[1];
    const float* logpsi = (const float*)d_in[2];
    float*       out    = (float*)d_out;

    hipLaunchKernelGGL(ricker_scan_kernel, dim3(1), dim3(32), 0, stream,
                       y0, params, logpsi, out);
}